// SimpleGCNLayer_66271345377741
// MI455X (gfx1250) — compile-verified
//
#include <hip/hip_runtime.h>
#include <hip/hip_bf16.h>
#include <stdint.h>

// ---------------------------------------------------------------------------
// GCNConv: out = D^{-1/2} (A+I) D^{-1/2} (x @ W) + b
//   x:[100000,128] f32, edge_index:[2,1600000] i64, W:[128,32] f32, b:[32] f32
// ---------------------------------------------------------------------------

#define D_IN  128
#define D_OUT 32

typedef __attribute__((ext_vector_type(2))) float v2f;
typedef __attribute__((ext_vector_type(8))) float v8f;

// ---- degree: init to 1.0 (self-loop), then count in-edges at dst ----------
__global__ __launch_bounds__(256) void gcn_deg_init(float* __restrict__ deg, int n) {
    int i = blockIdx.x * 256 + threadIdx.x;
    if (i < n) deg[i] = 1.0f;
}

__global__ __launch_bounds__(256) void gcn_deg_count(const long long* __restrict__ dst,
                                                     float* __restrict__ deg, int n_edges) {
    int e = blockIdx.x * 256 + threadIdx.x;
    if (e < n_edges) atomicAdd(&deg[(int)dst[e]], 1.0f);
}

__global__ __launch_bounds__(256) void gcn_deg_rsqrt(float* __restrict__ deg, int n) {
    int i = blockIdx.x * 256 + threadIdx.x;
    if (i < n) deg[i] = __frsqrt_rn(deg[i]);   // deg >= 1 always (self-loop)
}

// ---- h = x @ W via fp32 WMMA 16x16x4 --------------------------------------
// One wave computes a 16x32 output tile (two 16x16 accumulators sharing A).
// A 16x4 f32 layout: lanes 0-15 -> M=lane, K={k0,k0+1}; lanes 16-31 -> M=lane-16,
// K={k0+2,k0+3}; the per-lane pair is contiguous in x, so a single float2 load.
// B 4x16 f32 layout mirrors A with rows striped across lanes (N=lane&15).
__global__ __launch_bounds__(256) void gcn_gemm_wmma(const float* __restrict__ x,
                                                     const float* __restrict__ W,
                                                     float* __restrict__ h,
                                                     int n_nodes) {
    __shared__ float sW[D_IN * D_OUT];   // 16 KB of the 320 KB WGP LDS
    for (int i = threadIdx.x; i < D_IN * D_OUT; i += 256) sW[i] = W[i];
    __syncthreads();

    const int wave = blockIdx.x * (256 / 32) + (threadIdx.x >> 5);
    const int lane = threadIdx.x & 31;
    const int m0 = wave * 16;
    if (m0 >= n_nodes) return;          // wave-uniform: EXEC stays all-ones

    const int mrow  = m0 + (lane & 15);
    const int khalf = (lane >> 4) << 1; // 0 or 2
    const int n     = lane & 15;

    v8f acc0 = {};
    v8f acc1 = {};
    const float* xrow = x + (size_t)mrow * D_IN;

#pragma unroll 4
    for (int k0 = 0; k0 < D_IN; k0 += 4) {
        const int krow = k0 + khalf;
        v2f a = *(const v2f*)(xrow + krow);          // contiguous K pair
        v2f b0, b1;
        b0.x = sW[krow * D_OUT + n];                 // VGPR0 = even K
        b0.y = sW[(krow + 1) * D_OUT + n];           // VGPR1 = odd K
        b1.x = sW[krow * D_OUT + 16 + n];
        b1.y = sW[(krow + 1) * D_OUT + 16 + n];
        acc0 = __builtin_amdgcn_wmma_f32_16x16x4_f32(false, a, false, b0,
                                                     (short)0, acc0, false, false);
        acc1 = __builtin_amdgcn_wmma_f32_16x16x4_f32(false, a, false, b1,
                                                     (short)0, acc1, false, false);
    }

    // C/D 16x16 layout: lane L, VGPR i -> row = 8*(L>=16)+i, col = L&15
    const int rowoff = (lane >> 4) * 8;
    const int col    = lane & 15;
#pragma unroll
    for (int i = 0; i < 8; ++i) {
        float* orow = h + (size_t)(m0 + rowoff + i) * D_OUT;
        orow[col]      = acc0[i];
        orow[16 + col] = acc1[i];
    }
}

// ---- out = h * dinv^2 + b  (self-loop term + bias) ------------------------
__global__ __launch_bounds__(256) void gcn_self_bias(const float* __restrict__ h,
                                                     const float* __restrict__ dinv,
                                                     const float* __restrict__ bias,
                                                     float* __restrict__ out,
                                                     int n_nodes) {
    int idx = blockIdx.x * 256 + threadIdx.x;       // over n_nodes * 32
    int i = idx >> 5;
    int c = idx & 31;
    if (i < n_nodes) {
        float di = dinv[i];
        out[idx] = h[idx] * di * di + bias[c];
    }
}

// ---- edge scatter: 1 wave per edge, 1 lane per channel --------------------
// src/dst/dinv loads are wave-uniform (hardware broadcast); the h gather and
// the out atomic are each a single coalesced 128B transaction, L2-resident.
__global__ __launch_bounds__(256) void gcn_edge_scatter(const long long* __restrict__ src,
                                                        const long long* __restrict__ dst,
                                                        const float* __restrict__ h,
                                                        const float* __restrict__ dinv,
                                                        float* __restrict__ out,
                                                        int n_edges) {
    long long tid = (long long)blockIdx.x * 256 + threadIdx.x;
    int e = (int)(tid >> 5);                        // same edge across the wave32
    int c = (int)(tid & 31);
    if (e >= n_edges) return;
    int s = (int)src[e];
    int d = (int)dst[e];
    float norm = dinv[s] * dinv[d];
    float msg = h[(size_t)s * D_OUT + c] * norm;
    atomicAdd(&out[(size_t)d * D_OUT + c], msg);
}

// ---------------------------------------------------------------------------
extern "C" void kernel_launch(void* const* d_in, const int* in_sizes, int n_in,
                              void* d_out, int out_size, void* d_ws, size_t ws_size,
                              hipStream_t stream) {
    const float*     x    = (const float*)d_in[0];
    const long long* ei   = (const long long*)d_in[1];     // [2, E] int64
    const float*     W    = (const float*)d_in[2];
    const float*     bias = (const float*)d_in[3];
    float*           out  = (float*)d_out;

    const int n_nodes = in_sizes[0] / D_IN;
    const int n_edges = in_sizes[1] / 2;
    const long long* src = ei;
    const long long* dst = ei + n_edges;

    // workspace: h [n_nodes*32] f32, deg/dinv [n_nodes] f32
    float* h    = (float*)d_ws;
    float* dinv = h + (size_t)n_nodes * D_OUT;

    const int B = 256;
    int gN   = (n_nodes + B - 1) / B;
    int gE   = (n_edges + B - 1) / B;
    int tiles = (n_nodes + 15) / 16;
    int gG   = (tiles + (B / 32) - 1) / (B / 32);
    int gNC  = ((size_t)n_nodes * D_OUT + B - 1) / B;
    long long escatter = (long long)n_edges * 32;
    int gES  = (int)((escatter + B - 1) / B);

    gcn_deg_init   <<<gN,  B, 0, stream>>>(dinv, n_nodes);
    gcn_deg_count  <<<gE,  B, 0, stream>>>(dst, dinv, n_edges);
    gcn_deg_rsqrt  <<<gN,  B, 0, stream>>>(dinv, n_nodes);
    gcn_gemm_wmma  <<<gG,  B, 0, stream>>>(x, W, h, n_nodes);
    gcn_self_bias  <<<gNC, B, 0, stream>>>(h, dinv, bias, out, n_nodes);
    gcn_edge_scatter<<<gES, B, 0, stream>>>(src, dst, h, dinv, out, n_edges);
}